// AdDiffSortLoss_84138409329035
// MI455X (gfx1250) — compile-verified
//
#include <hip/hip_runtime.h>

#define TPB 256
#define N 8

typedef __attribute__((ext_vector_type(2))) float v2f;
typedef __attribute__((ext_vector_type(8))) float v8f;

// ---------- CDNA5 async global->LDS copy (ASYNCcnt path) ----------
__device__ __forceinline__ void async_ld_b128(unsigned lds_off, const void* gp) {
  // LDS destination address from lds_off; 16 bytes per active lane, tracked by ASYNCcnt
  asm volatile("global_load_async_to_lds_b128 %0, %1, off"
               :: "v"(lds_off), "v"(gp)
               : "memory");
}
__device__ __forceinline__ void wait_async0() {
  asm volatile("s_wait_asynccnt 0x0" ::: "memory");
}

// ---------- wave32 sum reduction through the matrix unit ----------
// A(16x4): VGPR0 = per-lane value v (rows M=lane%16, K=0 for lanes 0-15, K=2 for
// lanes 16-31), VGPR1 = 0. B = all-ones (layout-invariant). C = 0.
// D[m,n] = v[m] + v[m+16]; summing the 8 D VGPRs gives lanes 0-15 = sum(v[0:8])+
// sum(v[16:24]), lanes 16-31 = the complement; one xor-16 shuffle finishes.
// Requires EXEC all-ones (guaranteed: uniform control flow, full blocks).
__device__ __forceinline__ float wave_sum_wmma(float v) {
  v2f a; a.x = v;    a.y = 0.0f;
  v2f b; b.x = 1.0f; b.y = 1.0f;
  v8f c = {};
  v8f d = __builtin_amdgcn_wmma_f32_16x16x4_f32(
      /*neg_a=*/false, a, /*neg_b=*/false, b,
      /*c_mod=*/(short)0, c, /*reuse_a=*/false, /*reuse_b=*/false);
  float t = ((d[0] + d[1]) + (d[2] + d[3])) + ((d[4] + d[5]) + (d[6] + d[7]));
  t += __shfl_xor(t, 16, 32);
  return t;  // full-wave sum broadcast to every lane
}

// ---------- branchless atan-based sigmoid: alpha = atan(10*d)/pi + 0.5 ----------
__device__ __forceinline__ float atan_alpha(float d) {
  float t  = 10.0f * d;
  float at = __builtin_fabsf(t);
  float iv = __builtin_amdgcn_rcpf(at);     // v_rcp_f32
  bool  bg = at > 1.0f;
  float z  = bg ? iv : at;                  // z in [0,1]
  float z2 = z * z;
  float p  = __builtin_fmaf(z2, -0.01172120f, 0.05265332f);
  p = __builtin_fmaf(z2, p, -0.11643287f);
  p = __builtin_fmaf(z2, p,  0.19354346f);
  p = __builtin_fmaf(z2, p, -0.33262347f);
  p = __builtin_fmaf(z2, p,  0.99997726f);
  float r = z * p;
  r = bg ? (1.57079632679f - r) : r;
  r = __builtin_copysignf(r, t);
  return __builtin_fmaf(r, 0.31830988618f, 0.5f);   // atan/pi + 0.5
}

__device__ __forceinline__ float fast_ln(float v) {
  return 0.69314718056f * __builtin_amdgcn_logf(v); // ln2 * v_log_f32
}

template<int START>
__device__ __forceinline__ void sort_layer(float (&x)[N], float (&P)[N][N]) {
#pragma unroll
  for (int a = START; a + 1 < N; a += 2) {
    float av = x[a], bv = x[a + 1];
    float al = atan_alpha(bv - av);
    x[a]     = __builtin_fmaf(al, av - bv, bv);   // alpha*a + (1-alpha)*b
    x[a + 1] = __builtin_fmaf(al, bv - av, av);   // (1-alpha)*a + alpha*b
#pragma unroll
    for (int i = 0; i < N; ++i) {
      float A = P[i][a], B = P[i][a + 1];
      P[i][a]     = __builtin_fmaf(al, A - B, B);
      P[i][a + 1] = __builtin_fmaf(al, B - A, A);
    }
  }
}

__global__ __launch_bounds__(TPB)
void diffsort_ce_kernel(const float* __restrict__ pred,
                        const float* __restrict__ lab,
                        const float* __restrict__ ema,
                        float* __restrict__ partial,
                        int batch) {
  __shared__ float sPred[TPB * N];   // 8 KB
  __shared__ float sLab [TPB * N];   // 8 KB
  __shared__ float sEma [N];
  __shared__ float wred [TPB / 32];  // one slot per wave

  const int tid  = threadIdx.x;
  const int row  = blockIdx.x * TPB + tid;
  const int rowc = row < batch ? row : batch - 1;   // clamp: keep EXEC full & barriers uniform
  const float valid = (row < batch) ? 1.0f : 0.0f;

  // Stage this block's 256x8 tiles of pred & labels through the async copy engine.
  unsigned pB = (unsigned)(size_t)(&sPred[0]);
  unsigned lB = (unsigned)(size_t)(&sLab[0]);
  const char* gp = (const char*)(pred + (size_t)rowc * N);
  const char* gl = (const char*)(lab  + (size_t)rowc * N);
  async_ld_b128(pB + tid * 32u,       gp);
  async_ld_b128(pB + tid * 32u + 16u, gp + 16);
  async_ld_b128(lB + tid * 32u,       gl);
  async_ld_b128(lB + tid * 32u + 16u, gl + 16);
  if (tid < N) sEma[tid] = ema[tid];
  wait_async0();
  __syncthreads();

  float pr[N], lb[N];
#pragma unroll
  for (int i = 0; i < N; ++i) { pr[i] = sPred[tid * N + i]; lb[i] = sLab[tid * N + i]; }

  // rank_true[i] = stable descending rank of labels[i]
  int r[N];
#pragma unroll
  for (int i = 0; i < N; ++i) {
    int c = 0;
#pragma unroll
    for (int j = 0; j < N; ++j)
      c += (lb[j] > lb[i]) || ((lb[j] == lb[i]) && (j < i));
    r[i] = c;
  }

  // x = -(pred - rank_ema[rank_true])
  float x[N];
#pragma unroll
  for (int i = 0; i < N; ++i) x[i] = sEma[r[i]] - pr[i];

  // P = identity; run 8-layer odd-even network, column-blending P
  float P[N][N];
#pragma unroll
  for (int i = 0; i < N; ++i)
#pragma unroll
    for (int j = 0; j < N; ++j) P[i][j] = (i == j) ? 1.0f : 0.0f;

  sort_layer<0>(x, P); sort_layer<1>(x, P);
  sort_layer<0>(x, P); sort_layer<1>(x, P);
  sort_layer<0>(x, P); sort_layer<1>(x, P);
  sort_layer<0>(x, P); sort_layer<1>(x, P);

  // BCE sum: sum_ij log1p(-P) + per-row hot correction (hot col = rank_true[i])
  float s = 0.0f;
#pragma unroll
  for (int i = 0; i < N; ++i) {
    int   ri   = r[i];
    float hotP = P[i][0];
    float rsum = 0.0f;
#pragma unroll
    for (int j = 0; j < N; ++j) {
      float p = P[i][j];
      rsum += __builtin_fmaxf(fast_ln(1.0f - p), -100.0f);
      if (j > 0) hotP = (ri == j) ? p : hotP;
    }
    float lmHot = __builtin_fmaxf(fast_ln(1.0f - hotP), -100.0f);
    float lpHot = __builtin_fmaxf(fast_ln(hotP),        -100.0f);
    s += rsum - lmHot + lpHot;
  }
  s *= valid;

  // wave32 sum via the matrix unit, then ordered cross-wave combine (deterministic)
  float ws = wave_sum_wmma(s);
  if ((tid & 31) == 0) wred[tid >> 5] = ws;
  __syncthreads();
  if (tid == 0) {
    float tot = 0.0f;
#pragma unroll
    for (int w = 0; w < TPB / 32; ++w) tot += wred[w];
    partial[blockIdx.x] = tot;
  }
}

__global__ __launch_bounds__(TPB)
void finalize_kernel(const float* __restrict__ partial, float* __restrict__ out,
                     int nblk, float neg_scale) {
  __shared__ float wred[TPB / 32];
  const int tid = threadIdx.x;
  float s = 0.0f;
  for (int i = tid; i < nblk; i += TPB) s += partial[i];  // fixed per-lane order
  float ws = wave_sum_wmma(s);          // EXEC full: loop above is uniform-exit
  if ((tid & 31) == 0) wred[tid >> 5] = ws;
  __syncthreads();
  if (tid == 0) {
    float tot = 0.0f;
#pragma unroll
    for (int w = 0; w < TPB / 32; ++w) tot += wred[w];
    out[0] = tot * neg_scale;           // -mean over batch*64 entries
  }
}

extern "C" void kernel_launch(void* const* d_in, const int* in_sizes, int n_in,
                              void* d_out, int out_size, void* d_ws, size_t ws_size,
                              hipStream_t stream) {
  const float* pred = (const float*)d_in[0];
  const float* lab  = (const float*)d_in[1];
  const float* ema  = (const float*)d_in[2];
  float* out     = (float*)d_out;
  float* partial = (float*)d_ws;

  const int batch = in_sizes[0] / N;
  const int blocks = (batch + TPB - 1) / TPB;
  const float neg_scale = -1.0f / ((float)batch * (float)(N * N));

  diffsort_ce_kernel<<<blocks, TPB, 0, stream>>>(pred, lab, ema, partial, batch);
  finalize_kernel<<<1, TPB, 0, stream>>>(partial, out, blocks, neg_scale);
}